// CPABWarp_89378269430501
// MI455X (gfx1250) — compile-verified
//
#include <hip/hip_runtime.h>
#include <hip/hip_bf16.h>
#include <math.h>

// ---------------------------------------------------------------------------
// CPAB 1D warp:  pooled = mean(x, seq); ab = (B @ loc_w^T) @ pooled + B@loc_b;
// gamma = 50-step Euler integration of piecewise-affine velocity field.
//
// Shapes: input_seq (512,12,8192) f32, loc_w (12,15), loc_b (15), out (512,8192)
// ---------------------------------------------------------------------------

#define BATCH   512
#define CH      12
#define SEQ     8192
#define NCELLS  16
#define DTHETA  15          // NCELLS-1
#define NSTEPS  50
#define ROWS    (BATCH*CH)  // 6144

typedef float v2f __attribute__((ext_vector_type(2)));
typedef float v8f __attribute__((ext_vector_type(8)));

// ---------------------------------------------------------------------------
// Kernel A: mean over seq for each (batch, ch) row.  201 MB read -> HBM bound.
// One 256-thread block per row; 8 x float4 per thread, coalesced.
// ---------------------------------------------------------------------------
__global__ void cpab_pool_kernel(const float* __restrict__ in, float* __restrict__ pooled) {
    const int row = blockIdx.x;                       // 0..6143
    const float4* p = (const float4*)(in + (size_t)row * SEQ);
    float s = 0.0f;
    #pragma unroll
    for (int i = 0; i < 8; ++i) {
        float4 v = p[threadIdx.x + i * 256];          // 2048 float4 per row
        s += (v.x + v.y) + (v.z + v.w);
    }
    __shared__ float red[256];
    red[threadIdx.x] = s;
    __syncthreads();
    for (int off = 128; off > 0; off >>= 1) {
        if (threadIdx.x < off) red[threadIdx.x] += red[threadIdx.x + off];
        __syncthreads();
    }
    if (threadIdx.x == 0) pooled[row] = red[0] * (1.0f / (float)SEQ);
}

// ---------------------------------------------------------------------------
// Kernel B: build CPAB null-space basis B (32x15) via Householder QR of L^T,
// then fold with loc_w:  W2[i][c] = sum_t B[i][t]*loc_w[c][t]  (32x12),
// c0[i] = sum_t B[i][t]*loc_b[t].  One thread; tiny deterministic work.
// ---------------------------------------------------------------------------
__global__ void cpab_basis_kernel(const float* __restrict__ loc_w,
                                  const float* __restrict__ loc_b,
                                  float* __restrict__ W2, float* __restrict__ c0) {
    if (threadIdx.x != 0 || blockIdx.x != 0) return;

    const int R = DTHETA + 2;   // 17 constraints
    const int N = 2 * NCELLS;   // 32 unknowns

    double A[N][R];             // A = L^T (32 x 17)
    for (int i = 0; i < N; ++i)
        for (int j = 0; j < R; ++j) A[i][j] = 0.0;

    // Continuity at interior knots x_k = k/nC  (rows 0..14 of L)
    for (int k = 1; k < NCELLS; ++k) {
        double xk = (double)k / (double)NCELLS;
        A[2*(k-1)    ][k-1] =  xk;
        A[2*(k-1) + 1][k-1] =  1.0;
        A[2*k        ][k-1] = -xk;
        A[2*k + 1    ][k-1] = -1.0;
    }
    A[1][NCELLS-1] = 1.0;                 // v(0)=0  -> b_0 = 0
    A[2*NCELLS-2][NCELLS] = 1.0;          // v(1)=0  -> a_last + b_last = 0
    A[2*NCELLS-1][NCELLS] = 1.0;

    // Householder QR of A; store reflectors in V, squared norms in Wn.
    double V[R][N];
    double Wn[R];
    for (int j = 0; j < R; ++j) {
        double u[N];
        double nx2 = 0.0;
        for (int i = 0; i < N; ++i) { u[i] = (i >= j) ? A[i][j] : 0.0; nx2 += u[i]*u[i]; }
        double nx = sqrt(nx2);
        if (nx > 1e-300) {
            double sgn = (u[j] >= 0.0) ? 1.0 : -1.0;
            u[j] += sgn * nx;
            double un2 = 0.0;
            for (int i = 0; i < N; ++i) un2 += u[i]*u[i];
            for (int c = j; c < R; ++c) {
                double dot = 0.0;
                for (int i = j; i < N; ++i) dot += u[i]*A[i][c];
                double f = 2.0 * dot / un2;
                for (int i = j; i < N; ++i) A[i][c] -= f*u[i];
            }
            for (int i = 0; i < N; ++i) V[j][i] = u[i];
            Wn[j] = un2;
        } else {
            for (int i = 0; i < N; ++i) V[j][i] = 0.0;
            Wn[j] = 1.0;
        }
    }

    // Null-space basis columns = Q e_{R+t}, t = 0..14 : apply reflectors
    // in reverse to unit vectors.  Bm[i][t] = basis(32 x 15).
    double Bm[N][DTHETA];
    for (int t = 0; t < DTHETA; ++t) {
        double x[N];
        for (int i = 0; i < N; ++i) x[i] = 0.0;
        x[R + t] = 1.0;
        for (int j = R - 1; j >= 0; --j) {
            double dot = 0.0;
            for (int i = 0; i < N; ++i) dot += V[j][i]*x[i];
            double f = 2.0 * dot / Wn[j];
            for (int i = 0; i < N; ++i) x[i] -= f * V[j][i];
        }
        for (int i = 0; i < N; ++i) Bm[i][t] = x[i];
    }

    // Fold with localisation weights: W2 (32 x 12), c0 (32)
    for (int i = 0; i < N; ++i) {
        for (int c = 0; c < CH; ++c) {
            double acc = 0.0;
            for (int t = 0; t < DTHETA; ++t)
                acc += Bm[i][t] * (double)loc_w[c * DTHETA + t];
            W2[i * CH + c] = (float)acc;
        }
        double accb = 0.0;
        for (int t = 0; t < DTHETA; ++t) accb += Bm[i][t] * (double)loc_b[t];
        c0[i] = (float)accb;
    }
}

// ---------------------------------------------------------------------------
// Kernel C: AB[512][32] = pooled[512][12] @ W2^T[12][32] + c0, via
// V_WMMA_F32_16X16X4_F32.  One wave per 16x16 tile; K=12 -> 3 k-steps.
// grid = (M/16=32, N/16=2), block = 32 (one wave).
// A frag (16x4): lanes0-15 M=lane, VGPR0 K=k0+{0|2}, VGPR1 K=+1.
// B frag (4x16): lanes0-15 N=lane, VGPR0 K=k0+{0|2}, VGPR1 K=+1.
// C/D frag: VGPR v -> M = v + (lane<16?0:8), N = lane&15.
// ---------------------------------------------------------------------------
__global__ void cpab_param_gemm_kernel(const float* __restrict__ pooled,
                                       const float* __restrict__ W2,
                                       const float* __restrict__ c0,
                                       float* __restrict__ AB) {
    const int lane = threadIdx.x;         // 0..31
    const int mt = blockIdx.x;            // 0..31
    const int nt = blockIdx.y;            // 0..1
    const int half = (lane >= 16) ? 1 : 0;
    const int l15  = lane & 15;

    v8f acc;
    const float cini = c0[nt * 16 + l15];
    #pragma unroll
    for (int v = 0; v < 8; ++v) acc[v] = cini;

    #pragma unroll
    for (int k0 = 0; k0 < CH; k0 += 4) {
        const int kk = k0 + half * 2;
        v2f a, b;
        const int m = mt * 16 + l15;
        a.x = pooled[m * CH + kk];
        a.y = pooled[m * CH + kk + 1];
        const int n = nt * 16 + l15;
        b.x = W2[n * CH + kk];            // (W2^T)[kk][n]
        b.y = W2[n * CH + kk + 1];
        acc = __builtin_amdgcn_wmma_f32_16x16x4_f32(
            /*neg_a=*/false, a, /*neg_b=*/false, b,
            /*c_mod=*/(short)0, acc, /*reuse_a=*/false, /*reuse_b=*/false);
    }

    #pragma unroll
    for (int v = 0; v < 8; ++v) {
        const int m = mt * 16 + v + half * 8;
        const int n = nt * 16 + l15;
        AB[m * 32 + n] = acc[v];
    }
}

// ---------------------------------------------------------------------------
// Kernel D: 50-step Euler integration.  blockIdx.y = batch, 16-cell (a,b)
// LUT in LDS, 4 grid points per thread, fully register-resident phi.
// ---------------------------------------------------------------------------
__global__ void cpab_warp_kernel(const float* __restrict__ AB, float* __restrict__ out) {
    const int batch = blockIdx.y;
    __shared__ float ab[32];              // interleaved a_c, b_c
    if (threadIdx.x < 32) ab[threadIdx.x] = AB[batch * 32 + threadIdx.x];
    __syncthreads();

    const float dt = 1.0f / (float)NSTEPS;
    #pragma unroll
    for (int p = 0; p < 4; ++p) {
        const int gi = blockIdx.x * 1024 + p * 256 + threadIdx.x;
        float phi = (float)gi * (1.0f / (float)(SEQ - 1));   // linspace(0,1,SEQ)
        #pragma unroll 10
        for (int s = 0; s < NSTEPS; ++s) {
            int c = (int)floorf(phi * (float)NCELLS);
            c = (c < 0) ? 0 : ((c > NCELLS - 1) ? NCELLS - 1 : c);
            phi = phi + (ab[2*c] * phi + ab[2*c + 1]) * dt;
            phi = fminf(fmaxf(phi, 0.0f), 1.0f);
        }
        out[(size_t)batch * SEQ + gi] = phi;
    }
}

// ---------------------------------------------------------------------------
// Launch
// ---------------------------------------------------------------------------
extern "C" void kernel_launch(void* const* d_in, const int* in_sizes, int n_in,
                              void* d_out, int out_size, void* d_ws, size_t ws_size,
                              hipStream_t stream) {
    const float* input_seq = (const float*)d_in[0];
    // d_in[1] = mask (unused by reference math)
    const float* loc_w = (const float*)d_in[2];
    const float* loc_b = (const float*)d_in[3];
    // d_in[4] = n_cells (compile-time constant 16)

    float* ws     = (float*)d_ws;
    float* pooled = ws;                 // 6144 floats
    float* W2     = ws + ROWS;          // 384
    float* c0     = W2 + 32 * CH;       // 32
    float* AB     = c0 + 32;            // 16384
    float* out    = (float*)d_out;

    cpab_pool_kernel<<<ROWS, 256, 0, stream>>>(input_seq, pooled);
    cpab_basis_kernel<<<1, 1, 0, stream>>>(loc_w, loc_b, W2, c0);
    cpab_param_gemm_kernel<<<dim3(BATCH / 16, 2), 32, 0, stream>>>(pooled, W2, c0, AB);
    cpab_warp_kernel<<<dim3(SEQ / 1024, BATCH), 256, 0, stream>>>(AB, out);
}